// LocalAlignedFeatureFusion_36180804501805
// MI455X (gfx1250) — compile-verified
//
#include <hip/hip_runtime.h>

// Problem constants (from reference)
#define BB   2
#define CC   256
#define HH   128
#define WW   128
#define PP   (HH * WW)     // 16384
#define DD   64
#define KWIN 25
#define SCALEF 0.125f      // D^-0.5 = 64^-0.5

typedef __attribute__((ext_vector_type(2))) float v2f;
typedef __attribute__((ext_vector_type(8))) float v8f;
typedef __attribute__((ext_vector_type(4))) unsigned int v4u;
typedef __attribute__((ext_vector_type(8))) int v8i;
typedef __attribute__((ext_vector_type(4))) int v4i;

#define XS_STRIDE 33       // pad LDS rows to dodge bank conflicts (TDM pad does this)

// -----------------------------------------------------------------------------
// Tensor Data Mover: build D# per CDNA5 ISA §8.3-8.6 and issue TENSOR_LOAD_TO_LDS.
// Up to 3-D tile; tensor dims == tile dims (all our tiles are fully in-bounds).
//  g0: count=1 | lds_addr | global_addr[56:0] | type=2 (bits 127:126 = 10b)
//  g1: data_size=4B, pad ctrl, tensor_dim0/1, tile_dim0/1/2, dim0/dim1 strides
//  g2/g3: tensor_dim2 (rest zero / unused)
// Issued by one wave only (TDM ignores EXEC; all operands wave-uniform).
// -----------------------------------------------------------------------------
__device__ __forceinline__ void tdm_load_to_lds(
    unsigned int lds_off, const void* gptr,
    unsigned int tile0, unsigned int tile1, unsigned int tile2,
    unsigned long long stride0, unsigned long long stride1,
    unsigned int pad_en, unsigned int pad_iv, unsigned int pad_amt)
{
    const unsigned long long ga = (unsigned long long)(uintptr_t)gptr;
    v4u g0;
    g0.x = 1u;                                        // count=1 (valid user D#)
    g0.y = lds_off;                                   // LDS byte address
    g0.z = (unsigned int)ga;                          // global_addr[31:0]
    g0.w = ((unsigned int)(ga >> 32) & 0x01FFFFFFu)   // global_addr[56:32]
           | 0x80000000u;                             // type=2 ("image")

    v8i g1;
    g1[0] = (int)((2u << 16)                          // data_size = 4 bytes
                  | (pad_en << 20) | (pad_iv << 22) | (pad_amt << 25));
    g1[1] = (int)((tile0 & 0xFFFFu) << 16);           // tensor_dim0[15:0] (== tile0)
    g1[2] = (int)(((tile0 >> 16) & 0xFFFFu)           // tensor_dim0[31:16]
                  | ((tile1 & 0xFFFFu) << 16));       // tensor_dim1[15:0]
    g1[3] = (int)(((tile1 >> 16) & 0xFFFFu)           // tensor_dim1[31:16]
                  | ((tile0 & 0xFFFFu) << 16));       // tile_dim0
    g1[4] = (int)((tile1 & 0xFFFFu)                   // tile_dim1
                  | ((tile2 & 0xFFFFu) << 16));       // tile_dim2 (0 => 2-D)
    g1[5] = (int)(unsigned int)stride0;               // tensor_dim0_stride[31:0]
    g1[6] = (int)((unsigned int)((stride0 >> 32) & 0xFFFFu)
                  | ((unsigned int)(stride1 & 0xFFFFu) << 16));
    g1[7] = (int)(unsigned int)((stride1 >> 16) & 0xFFFFFFFFu);

    v4i g2;                                           // tensor_dim2, rest zero
    g2[0] = (int)tile2; g2[1] = 0; g2[2] = 0; g2[3] = 0;
    v4i g3; g3[0] = 0; g3[1] = 0; g3[2] = 0; g3[3] = 0;

#if defined(__clang_major__) && __clang_major__ >= 23
    v8i g4; g4[0]=0; g4[1]=0; g4[2]=0; g4[3]=0; g4[4]=0; g4[5]=0; g4[6]=0; g4[7]=0;
    __builtin_amdgcn_tensor_load_to_lds(g0, g1, g2, g3, g4, 0);
#else
    __builtin_amdgcn_tensor_load_to_lds(g0, g1, g2, g3, 0);
#endif
}

// -----------------------------------------------------------------------------
// Kernel 1: Q/K projection GEMM with fp32 WMMA (16x16x4).
//   out[d, p] = sum_c W[d, c] * X[c, p] + bias[d]
// grid = (P/32, 2 (q|k), B), block = 256 (8 waves).
// Input tile staged by TDM with native LDS padding (1 DWORD per 32 => stride 33).
// Q/K written pixel-major [B][P][D] so consumers get contiguous 64-float rows.
// -----------------------------------------------------------------------------
__global__ __launch_bounds__(256) void proj_qk_kernel(
    const float* __restrict__ anchor, const float* __restrict__ source,
    const float* __restrict__ Wq, const float* __restrict__ bq,
    const float* __restrict__ Wk, const float* __restrict__ bk,
    float* __restrict__ qws, float* __restrict__ kws)
{
    const int tid  = threadIdx.x;
    const int lane = tid & 31;
    const int wave = tid >> 5;            // 0..7
    const int p0   = blockIdx.x * 32;     // pixel tile base
    const int sel  = blockIdx.y;          // 0 = Q (anchor), 1 = K (source)
    const int b    = blockIdx.z;

    const float* x    = (sel == 0 ? anchor : source) + (size_t)b * CC * PP;
    const float* wmat = (sel == 0 ? Wq : Wk);
    const float* bias = (sel == 0 ? bq : bk);
    float*       outp = (sel == 0 ? qws : kws) + (size_t)b * PP * DD;

    // X tile [256 c][32 px] in LDS, stride 33, loaded by the Tensor Data Mover:
    // 2-D tile 32x256, row stride P, pad 1 DWORD (pad_amt=0) every 32 DWORDs
    // (pad_iv=4 => 1<<4 * 8B == 128B == 32 DWORDs).
    __shared__ float xs[CC * XS_STRIDE];
    if (wave == 0) {
        tdm_load_to_lds((unsigned int)(uintptr_t)&xs[0], x + p0,
                        /*tile0=*/32, /*tile1=*/CC, /*tile2=*/0,
                        /*stride0=*/PP, /*stride1=*/0,
                        /*pad_en=*/1, /*pad_iv=*/4, /*pad_amt=*/0);
        __builtin_amdgcn_s_wait_tensorcnt(0);
    }
    __syncthreads();

    const int half = lane >> 4;                 // 0: K {k,k+1}, 1: K {k+2,k+3}
    const int l15  = lane & 15;
    const int mrow = (wave & 3) * 16 + l15;     // A-matrix row (output d)
    const int ncol = (wave >> 2) * 16 + l15;    // B-matrix col (pixel in tile)

    v8f acc = {};
#pragma unroll 4
    for (int k = 0; k < CC; k += 4) {
        const int ka = k + half * 2;
        v2f a, bf;
        // A (16x4 f32): lane row = mrow; VGPR0/1 hold K = ka / ka+1.
        a.x = wmat[mrow * CC + ka];
        a.y = wmat[mrow * CC + ka + 1];
        // B (4x16 f32): row K striped across lanes; VGPR0/1 hold K = ka / ka+1.
        bf.x = xs[ka * XS_STRIDE + ncol];
        bf.y = xs[(ka + 1) * XS_STRIDE + ncol];
        acc = __builtin_amdgcn_wmma_f32_16x16x4_f32(
            /*neg_a=*/false, a, /*neg_b=*/false, bf,
            /*c_mod=*/(short)0, acc, /*reuse_a=*/false, /*reuse_b=*/false);
    }

    // C/D 16x16 f32 layout: VGPR j holds row (j + half*8), col = l15.
    const int d0  = (wave & 3) * 16 + half * 8;
    const int col = p0 + ncol;
    float* dst = outp + (size_t)col * DD + d0;   // contiguous 8 floats, 32B aligned
#pragma unroll
    for (int j = 0; j < 8; ++j)
        dst[j] = acc[j] + bias[d0 + j];
}

// -----------------------------------------------------------------------------
// 25-tap weighted aggregation over one 8-channel LDS tile (6 rows x 128 cols).
// -----------------------------------------------------------------------------
__device__ __forceinline__ void agg_compute(
    const float* __restrict__ tile, const float* __restrict__ wgt,
    int xq, int ly, int b, int c0, int p, float* __restrict__ out)
{
#pragma unroll
    for (int ch = 0; ch < 8; ++ch) {
        float acc = 0.0f;
#pragma unroll
        for (int kk = 0; kk < KWIN; ++kk) {
            const int dy = kk / 5 - 2, dx = kk % 5 - 2;
            int rxx = xq + dx;
            rxx = rxx < 0 ? 0 : (rxx > WW - 1 ? WW - 1 : rxx); // safe addr; wgt=0 if OOB
            acc += wgt[kk] * tile[ch * (6 * WW) + ((ly + dy) << 7) + rxx];
        }
        out[(size_t)(b * CC + c0 + ch) * PP + p] = acc;
    }
}

// -----------------------------------------------------------------------------
// Kernel 2: fused scores -> softmax -> V aggregation.
// One thread per pixel; block = 256 threads = 2 full image rows.
// Interior blocks: double-buffered TDM pipeline (DMA of chunk i+1 overlaps the
// aggregation of chunk i). Edge blocks (need zero rows): manual fill.
// -----------------------------------------------------------------------------
__global__ __launch_bounds__(256) void attn_fuse_kernel(
    const float* __restrict__ source,
    const float* __restrict__ qws, const float* __restrict__ kws,
    float* __restrict__ out)
{
    const int idx = blockIdx.x * 256 + threadIdx.x;
    const int b   = idx >> 14;          // / P
    const int p   = idx & (PP - 1);
    const int y   = p >> 7;             // / W
    const int xq  = p & (WW - 1);

    // --- load this pixel's q vector (contiguous 64 floats) ---
    float q[DD];
    {
        const float4* q4 = (const float4*)(qws + ((size_t)b * PP + p) * DD);
#pragma unroll
        for (int i = 0; i < 16; ++i) {
            float4 t = q4[i];
            q[4 * i + 0] = t.x; q[4 * i + 1] = t.y;
            q[4 * i + 2] = t.z; q[4 * i + 3] = t.w;
        }
    }

    // --- 25 attention logits (0 for out-of-bounds taps, matching zero-pad) ---
    float sc[KWIN];
#pragma unroll
    for (int kk = 0; kk < KWIN; ++kk) {
        const int dy = kk / 5 - 2, dx = kk % 5 - 2;
        const int yy = y + dy, xx = xq + dx;
        const bool ok = (yy >= 0) && (yy < HH) && (xx >= 0) && (xx < WW);
        float s = 0.0f;
        if (ok) {
            const float4* k4 =
                (const float4*)(kws + ((size_t)b * PP + yy * WW + xx) * DD);
#pragma unroll
            for (int i = 0; i < 16; ++i) {
                float4 t = k4[i];
                s += q[4 * i + 0] * t.x + q[4 * i + 1] * t.y +
                     q[4 * i + 2] * t.z + q[4 * i + 3] * t.w;
            }
            s *= SCALEF;
        }
        sc[kk] = s;
    }

    // --- softmax over 25 (OOB taps kept with logit 0, as in reference) ---
    float m = sc[0];
#pragma unroll
    for (int kk = 1; kk < KWIN; ++kk) m = fmaxf(m, sc[kk]);
    float wgt[KWIN], sum = 0.0f;
#pragma unroll
    for (int kk = 0; kk < KWIN; ++kk) { wgt[kk] = __expf(sc[kk] - m); sum += wgt[kk]; }
    const float inv = 1.0f / sum;
#pragma unroll
    for (int kk = 0; kk < KWIN; ++kk) {
        const int dy = kk / 5 - 2, dx = kk % 5 - 2;
        const int yy = y + dy, xx = xq + dx;
        const bool ok = (yy >= 0) && (yy < HH) && (xx >= 0) && (xx < WW);
        wgt[kk] = ok ? wgt[kk] * inv : 0.0f;   // OOB tap: V is zero anyway
    }

    // --- V aggregation: 8 channels x 6 rows x 128 cols LDS tiles ---
    __shared__ float smem[2][8 * 6 * WW];      // 2 x 24 KB (double buffer)
    const int y0 = ((blockIdx.x * 256) & (PP - 1)) >> 7;   // first of 2 rows
    const int ly = y - (y0 - 2);                            // 2 or 3
    const float* srcb = source + (size_t)b * CC * PP;
    const bool interior = (y0 >= 2) && (y0 + 3 < HH);

    if (interior) {
        // TDM double-buffered pipeline: 3-D tile (x=128, rows=6 stride W,
        // channels=8 stride P), contiguous into LDS, one wave issues.
        const float* t0 = srcb + (size_t)(y0 - 2) * WW;    // chunk base (c=0)
        if (threadIdx.x < 32)
            tdm_load_to_lds((unsigned int)(uintptr_t)&smem[0][0], t0,
                            /*tile0=*/WW, /*tile1=*/6, /*tile2=*/8,
                            /*stride0=*/WW, /*stride1=*/PP,
                            /*pad_en=*/0, /*pad_iv=*/0, /*pad_amt=*/0);
        for (int c0 = 0; c0 < CC; c0 += 8) {
            const int ib = (c0 >> 3) & 1;
            if (threadIdx.x < 32) __builtin_amdgcn_s_wait_tensorcnt(0);
            __syncthreads();   // chunk c0 resident; buf[ib^1] consumers done
            if ((threadIdx.x < 32) && (c0 + 8 < CC))
                tdm_load_to_lds((unsigned int)(uintptr_t)&smem[ib ^ 1][0],
                                t0 + (size_t)(c0 + 8) * PP,
                                WW, 6, 8, WW, PP, 0, 0, 0);
            agg_compute(&smem[ib][0], wgt, xq, ly, b, c0, p, out);
            __syncthreads();   // done reading buf[ib] before it is refilled
        }
    } else {
        // Edge rows: manual fill with zero padding for out-of-image rows.
        for (int c0 = 0; c0 < CC; c0 += 8) {
            __syncthreads();
#pragma unroll
            for (int i = 0; i < 24; ++i) {
                int e  = i * 256 + threadIdx.x;
                int ch = e / (6 * WW);
                int r  = (e % (6 * WW)) >> 7;
                int cx = e & (WW - 1);
                int yy = y0 - 2 + r;
                float v = (yy >= 0 && yy < HH)
                          ? srcb[(size_t)(c0 + ch) * PP + yy * WW + cx] : 0.0f;
                smem[0][ch * (6 * WW) + (r << 7) + cx] = v;
            }
            __syncthreads();
            agg_compute(&smem[0][0], wgt, xq, ly, b, c0, p, out);
        }
    }
}

// -----------------------------------------------------------------------------
extern "C" void kernel_launch(void* const* d_in, const int* in_sizes, int n_in,
                              void* d_out, int out_size, void* d_ws, size_t ws_size,
                              hipStream_t stream)
{
    const float* anchor = (const float*)d_in[0];
    const float* source = (const float*)d_in[1];
    const float* Wq     = (const float*)d_in[2];
    const float* bq     = (const float*)d_in[3];
    const float* Wk     = (const float*)d_in[4];
    const float* bk     = (const float*)d_in[5];
    float* out = (float*)d_out;

    float* qws = (float*)d_ws;                         // [B][P][D]
    float* kws = qws + (size_t)BB * PP * DD;           // [B][P][D] (total 16.8 MB)

    dim3 g1(PP / 32, 2, BB);
    proj_qk_kernel<<<g1, 256, 0, stream>>>(anchor, source, Wq, bq, Wk, bk, qws, kws);

    attn_fuse_kernel<<<(BB * PP) / 256, 256, 0, stream>>>(source, qws, kws, out);
}